// GraphSAGE_9689446219933
// MI455X (gfx1250) — compile-verified
//
#include <hip/hip_runtime.h>

// GraphSAGE (mean) 2-layer forward for MI455X / gfx1250.
// N=100000 nodes, E=800000 edges, D=128 -> H=256 -> H=256, fp32 throughout.
//
// Strategy (roofline-driven):
//  * Avoid the 307M fp32 scatter-atomics of the naive approach (atomic-op
//    throughput bound). Instead build a dst-CSR per call (int histogram +
//    LDS block scan + cursor fill), then aggregate by GATHER: one wave per
//    node, float4 lanes, coalesced global_load_b128, zero float atomics.
//  * GEMMs (39 GFLOP total) use V_WMMA_F32_16X16X4_F32 at full fp32
//    precision; each wave owns a 16x64 output strip (4 x v8f accumulators,
//    A-fragment reused 4x, weights stay resident in L2).
//  * All control flow is wave-uniform -> EXEC all-ones as WMMA requires.

typedef __attribute__((ext_vector_type(2))) float v2f;
typedef __attribute__((ext_vector_type(4))) float v4f;
typedef __attribute__((ext_vector_type(8))) float v8f;

static constexpr int HOUT = 256;

__device__ __forceinline__ int atomicAddIntRelaxed(int* p, int v) {
  return __hip_atomic_fetch_add(p, v, __ATOMIC_RELAXED, __HIP_MEMORY_SCOPE_AGENT);
}

// ---------------------------------------------------------------- utilities
__global__ void zero_i32_kernel(int* __restrict__ p, int n) {
  int i = blockIdx.x * blockDim.x + threadIdx.x;
  int stride = gridDim.x * blockDim.x;
  for (; i < n; i += stride) p[i] = 0;
}

// -------------------------------------------------------------- CSR build
__global__ void count_kernel(const int* __restrict__ dst, int* __restrict__ degcnt, int E) {
  int e = blockIdx.x * blockDim.x + threadIdx.x;
  if (e < E) atomicAddIntRelaxed(&degcnt[dst[e]], 1);
}

// Block-level exclusive scan: 256 threads x 4 items = 1024 elements/block.
__global__ void scan1_kernel(const int* __restrict__ in, int* __restrict__ out,
                             int* __restrict__ blocksums, int n) {
  __shared__ int sh[256];
  const int t = threadIdx.x;
  const int base = blockIdx.x * 1024 + t * 4;
  int v0 = (base + 0 < n) ? in[base + 0] : 0;
  int v1 = (base + 1 < n) ? in[base + 1] : 0;
  int v2 = (base + 2 < n) ? in[base + 2] : 0;
  int v3 = (base + 3 < n) ? in[base + 3] : 0;
  const int tsum = v0 + v1 + v2 + v3;
  sh[t] = tsum;
  __syncthreads();
  for (int off = 1; off < 256; off <<= 1) {
    int x = (t >= off) ? sh[t - off] : 0;
    __syncthreads();
    sh[t] += x;
    __syncthreads();
  }
  const int excl = sh[t] - tsum;
  if (t == 255) blocksums[blockIdx.x] = sh[255];
  if (base + 0 < n) out[base + 0] = excl;
  if (base + 1 < n) out[base + 1] = excl + v0;
  if (base + 2 < n) out[base + 2] = excl + v0 + v1;
  if (base + 3 < n) out[base + 3] = excl + v0 + v1 + v2;
}

// Serial exclusive scan of the (<=1024) block sums; tiny.
__global__ void scan2_kernel(int* __restrict__ blocksums, int nb) {
  if (blockIdx.x == 0 && threadIdx.x == 0) {
    int acc = 0;
    for (int i = 0; i < nb; ++i) {
      int t = blocksums[i];
      blocksums[i] = acc;
      acc += t;
    }
  }
}

// Add block offsets; also seed the fill cursors (cursor becomes rowend after fill).
__global__ void scan3_kernel(int* __restrict__ rowptr, const int* __restrict__ blocksums,
                             int* __restrict__ cursor, int n) {
  int i = blockIdx.x * blockDim.x + threadIdx.x;
  if (i < n) {
    int v = rowptr[i] + blocksums[i >> 10];
    rowptr[i] = v;
    cursor[i] = v;
  }
}

__global__ void fill_kernel(const int* __restrict__ src, const int* __restrict__ dst,
                            int* __restrict__ cursor, int* __restrict__ col, int E) {
  int e = blockIdx.x * blockDim.x + threadIdx.x;
  if (e < E) {
    int pos = atomicAddIntRelaxed(&cursor[dst[e]], 1);
    col[pos] = src[e];
  }
}

// ------------------------------------------------- mean aggregation (gather)
// One wave per node. D=128: each lane owns one float4 slice (lane*4).
// D=256: each lane owns two float4 slices (lane*4 and 128+lane*4).
// No float atomics; all loads are coalesced global_load_b128.
template <int D>
__global__ void __launch_bounds__(256)
gather_mean_kernel(const float* __restrict__ X, const int* __restrict__ col,
                   const int* __restrict__ rowstart, const int* __restrict__ rowend,
                   float* __restrict__ Mean, int N) {
  const int wave = (blockIdx.x * blockDim.x + threadIdx.x) >> 5;
  const int lane = threadIdx.x & 31;
  if (wave >= N) return;  // wave-uniform

  const int start = rowstart[wave];
  const int end = rowend[wave];

  v4f acc0 = {0.0f, 0.0f, 0.0f, 0.0f};
  v4f acc1 = {0.0f, 0.0f, 0.0f, 0.0f};

  int j = start;
  for (; j + 2 <= end; j += 2) {  // unroll x2 for latency hiding
    const int u0 = col[j];
    const int u1 = col[j + 1];
    const v4f* p0 = (const v4f*)(X + (long)u0 * D);
    const v4f* p1 = (const v4f*)(X + (long)u1 * D);
    acc0 += p0[lane];
    acc0 += p1[lane];
    if (D == 256) {
      acc1 += p0[lane + 32];
      acc1 += p1[lane + 32];
    }
  }
  if (j < end) {
    const int u0 = col[j];
    const v4f* p0 = (const v4f*)(X + (long)u0 * D);
    acc0 += p0[lane];
    if (D == 256) acc1 += p0[lane + 32];
  }

  const float s = 1.0f / fmaxf((float)(end - start), 1.0f);
  v4f* mp = (v4f*)(Mean + (long)wave * D);
  mp[lane] = acc0 * s;
  if (D == 256) mp[lane + 32] = acc1 * s;
}

// ------------------------------------------------------------- fused GEMM
// Out[M, 256] = act( Aself @ Wself + Amean @ Wneigh + bias )
// Per wave: 16(M) x 64(N) strip -> 4 x v8f accumulators; block = 8 waves =
// 2 M-tiles x full 256-column width. Requires M % 32 == 0 (100000 = 3125*32).
template <int K, bool RELU>
__global__ void __launch_bounds__(256)
sage_gemm_kernel(const float* __restrict__ Aself, const float* __restrict__ Amean,
                 const float* __restrict__ Wself, const float* __restrict__ Wneigh,
                 const float* __restrict__ bias, float* __restrict__ Out) {
  const int lane  = threadIdx.x & 31;
  const int wave  = threadIdx.x >> 5;                 // 0..7
  const int m0    = blockIdx.x * 32 + (wave >> 2) * 16;
  const int nbase = (wave & 3) * 64;
  const int l15   = lane & 15;
  const int kh    = (lane >> 4) * 2;                  // lanes 0-15: K+0/1, lanes 16-31: K+2/3
  const int arow  = m0 + l15;

  v8f c0 = {}, c1 = {}, c2 = {}, c3 = {};

  for (int pass = 0; pass < 2; ++pass) {
    const float* A = pass ? Amean : Aself;
    const float* W = pass ? Wneigh : Wself;
    const float* ap = A + (long)arow * K + kh;        // 8-byte aligned (kh even)

    for (int k = 0; k < K; k += 4) {
      // A fragment (16x4 f32): direct 8-byte load
      v2f a = *(const v2f*)(ap + k);

      // B fragments (4x16 f32): vgpr0 = row k+kh, vgpr1 = row k+kh+1
      const float* w0 = W + (long)(k + kh) * HOUT + nbase + l15;
      const float* w1 = w0 + HOUT;
      v2f b;
      b.x = w0[0];  b.y = w1[0];
      c0 = __builtin_amdgcn_wmma_f32_16x16x4_f32(false, a, false, b, (short)0, c0, false, false);
      b.x = w0[16]; b.y = w1[16];
      c1 = __builtin_amdgcn_wmma_f32_16x16x4_f32(false, a, false, b, (short)0, c1, false, false);
      b.x = w0[32]; b.y = w1[32];
      c2 = __builtin_amdgcn_wmma_f32_16x16x4_f32(false, a, false, b, (short)0, c2, false, false);
      b.x = w0[48]; b.y = w1[48];
      c3 = __builtin_amdgcn_wmma_f32_16x16x4_f32(false, a, false, b, (short)0, c3, false, false);
    }
  }

  // C/D layout: vgpr i -> row m0 + i + 8*(lane>>4), col n0 + l15
  const int rbase = m0 + 8 * (lane >> 4);
#define SAGE_STORE_TILE(CT, T)                                              \
  do {                                                                      \
    const int n0 = nbase + 16 * (T);                                        \
    const float bv = bias[n0 + l15];                                        \
    _Pragma("unroll") for (int i = 0; i < 8; ++i) {                         \
      float v = (CT)[i] + bv;                                               \
      if (RELU) v = fmaxf(v, 0.0f);                                         \
      Out[(long)(rbase + i) * HOUT + n0 + l15] = v;                         \
    }                                                                       \
  } while (0)
  SAGE_STORE_TILE(c0, 0);
  SAGE_STORE_TILE(c1, 1);
  SAGE_STORE_TILE(c2, 2);
  SAGE_STORE_TILE(c3, 3);
#undef SAGE_STORE_TILE
}

// ------------------------------------------------------------------ driver
static inline size_t align_up(size_t x, size_t a) { return (x + a - 1) & ~(a - 1); }

extern "C" void kernel_launch(void* const* d_in, const int* in_sizes, int n_in,
                              void* d_out, int out_size, void* d_ws, size_t ws_size,
                              hipStream_t stream) {
  const float* feat     = (const float*)d_in[0];
  const int*   src      = (const int*)d_in[1];
  const int*   dst      = (const int*)d_in[2];
  const float* w_self1  = (const float*)d_in[3];
  const float* w_neigh1 = (const float*)d_in[4];
  const float* b1       = (const float*)d_in[5];
  const float* w_self2  = (const float*)d_in[6];
  const float* w_neigh2 = (const float*)d_in[7];
  const float* b2       = (const float*)d_in[8];

  const int D = 128, H = 256;
  const int N = in_sizes[0] / D;   // 100000
  const int E = in_sizes[1];       // 800000
  const int nb = (N + 1023) / 1024;

  // Workspace layout (1 KB aligned):
  //   degcnt[N] | rowptr[N] | cursor[N] | blocksums[nb] | col[E]   (ints)
  //   mean1[N*128] | h1[N*256] | mean2[N*256]                      (floats)
  char* base = (char*)d_ws;
  int* degcnt    = (int*)base;  base += align_up((size_t)N * 4, 1024);
  int* rowptr    = (int*)base;  base += align_up((size_t)N * 4, 1024);
  int* cursor    = (int*)base;  base += align_up((size_t)N * 4, 1024);
  int* blocksums = (int*)base;  base += align_up((size_t)nb * 4, 1024);
  int* col       = (int*)base;  base += align_up((size_t)E * 4, 1024);
  float* mean1   = (float*)base; base += align_up((size_t)N * D * 4, 1024);
  float* h1      = (float*)base; base += align_up((size_t)N * H * 4, 1024);
  float* mean2   = (float*)base;

  // --- CSR build (per call; int atomics only) ---
  zero_i32_kernel<<<512, 256, 0, stream>>>(degcnt, N);
  count_kernel<<<(E + 255) / 256, 256, 0, stream>>>(dst, degcnt, E);
  scan1_kernel<<<nb, 256, 0, stream>>>(degcnt, rowptr, blocksums, N);
  scan2_kernel<<<1, 64, 0, stream>>>(blocksums, nb);
  scan3_kernel<<<(N + 255) / 256, 256, 0, stream>>>(rowptr, blocksums, cursor, N);
  fill_kernel<<<(E + 255) / 256, 256, 0, stream>>>(src, dst, cursor, col, E);
  // cursor[v] now equals row end of v.

  // --- Layer 1: gather-mean, then WMMA GEMM + bias + ReLU -> h1 ---
  gather_mean_kernel<128><<<(N + 7) / 8, 256, 0, stream>>>(feat, col, rowptr, cursor, mean1, N);
  sage_gemm_kernel<128, true><<<N / 32, 256, 0, stream>>>(
      feat, mean1, w_self1, w_neigh1, b1, h1);

  // --- Layer 2: gather-mean on h1, then WMMA GEMM + bias -> out ---
  gather_mean_kernel<256><<<(N + 7) / 8, 256, 0, stream>>>(h1, col, rowptr, cursor, mean2, N);
  sage_gemm_kernel<256, false><<<N / 32, 256, 0, stream>>>(
      h1, mean2, w_self2, w_neigh2, b2, (float*)d_out);

  (void)n_in; (void)out_size; (void)ws_size;
}